// GE2ELoss_32925219291410
// MI455X (gfx1250) — compile-verified
//
#include <hip/hip_runtime.h>

typedef __attribute__((ext_vector_type(2))) float v2f;
typedef __attribute__((ext_vector_type(8))) float v8f;

#define NROWS 65536
#define DDIM  512
#define KCLS  512

// ---------------------------------------------------------------- utilities
__global__ void k_zero(float* __restrict__ p, int n) {
  int i = blockIdx.x * blockDim.x + threadIdx.x;
  if (i < n) p[i] = 0.0f;
}

// Fused: inv_norm[i] = 1/max(||emb_i||,1e-12)  AND  segment-sum of normalized
// rows via f32 global atomics. One wave per row; the 16 row elements per lane
// stay in registers across both phases (one HBM pass instead of two).
__global__ void k_norm_accum(const float* __restrict__ emb, const int* __restrict__ labels,
                             float* __restrict__ inv_norm,
                             float* __restrict__ sums, float* __restrict__ counts) {
  int row  = (blockIdx.x * blockDim.x + threadIdx.x) >> 5;
  int lane = threadIdx.x & 31;
  const float* r = emb + (size_t)row * DDIM;
  float x[DDIM / 32];
  float ss = 0.0f;
#pragma unroll
  for (int j = 0; j < DDIM / 32; ++j) {
    x[j] = r[lane + 32 * j];
    ss += x[j] * x[j];
  }
#pragma unroll
  for (int m = 16; m > 0; m >>= 1) ss += __shfl_xor(ss, m, 32);
  float s = 1.0f / fmaxf(sqrtf(ss), 1e-12f);
  if (lane == 0) inv_norm[row] = s;
  int l = labels[row];
  float* dst = sums + (size_t)l * DDIM;
#pragma unroll
  for (int j = 0; j < DDIM / 32; ++j) {
    atomicAdd(&dst[lane + 32 * j], x[j] * s);
  }
  if (lane == 0) atomicAdd(&counts[l], 1.0f);
}

// ctrT[d][k] = sums[k][d]/max(counts[k],1) ; snorm2[k] = ||sums[k]||^2
__global__ void k_centroid(const float* __restrict__ sums, const float* __restrict__ counts,
                           float* __restrict__ ctrT, float* __restrict__ snorm2) {
  int k   = blockIdx.x;     // KCLS blocks
  int tid = threadIdx.x;    // 256 threads
  float inv = 1.0f / fmaxf(counts[k], 1.0f);
  float sq  = 0.0f;
#pragma unroll
  for (int j = 0; j < DDIM / 256; ++j) {
    int d   = tid + 256 * j;
    float v = sums[(size_t)k * DDIM + d];
    ctrT[(size_t)d * KCLS + k] = v * inv;
    sq += v * v;
  }
#pragma unroll
  for (int m = 16; m > 0; m >>= 1) sq += __shfl_xor(sq, m, 32);
  __shared__ float red[8];
  int lane = tid & 31, wid = tid >> 5;
  if (lane == 0) red[wid] = sq;
  __syncthreads();
  if (tid == 0) {
    float t = 0.0f;
#pragma unroll
    for (int w = 0; w < 8; ++w) t += red[w];
    snorm2[k] = t;
  }
}

// ------------------- fused GEMM (fp32 WMMA) + own-substitution + log-softmax
// block = 256 thr (8 waves); each wave owns 16 rows x all 512 cols (32 C tiles)
// Register-pipelined staging: global loads for chunk ch+1 issue before the
// WMMA burst for chunk ch; LDS commit happens after the next barrier.
#define DC 16   // D-chunk staged in LDS per iteration
__global__ __launch_bounds__(256)
void k_gemm_loss(const float* __restrict__ emb, const int* __restrict__ labels,
                 const float* __restrict__ inv_norm, const float* __restrict__ ctrT,
                 const float* __restrict__ counts, const float* __restrict__ snorm2,
                 const float* __restrict__ wptr, const float* __restrict__ bptr,
                 float* __restrict__ loss_acc) {
  __shared__ float A_lds[128 * DC];   // [row][dl]  8 KB  (pre-scaled by inv_norm)
  __shared__ float B_lds[KCLS * DC];  // [k][dl]   32 KB  (centroids, k-major)
  __shared__ float red[8];

  const int tid  = threadIdx.x;
  const int lane = tid & 31;
  const int wid  = tid >> 5;
  const int lo   = lane & 15;
  const int hi   = lane >> 4;
  const int row0 = blockIdx.x * 128;

  v8f zero = {};
  v8f c[32];
#pragma unroll
  for (int t = 0; t < 32; ++t) c[t] = zero;

  // staging-thread geometry (uniform across waves)
  const int rA    = tid >> 2;          // 0..63, two passes of 64 rows
  const int partA = tid & 3;
  const float sc0 = inv_norm[row0 + rA];
  const float sc1 = inv_norm[row0 + rA + 64];

  float4 aR[2];
  float4 bR[8];

  // issue global loads for a chunk into registers
  auto load_chunk = [&](int d0) {
    float4 v0 = *(const float4*)&emb[(size_t)(row0 + rA) * DDIM + d0 + partA * 4];
    v0.x *= sc0; v0.y *= sc0; v0.z *= sc0; v0.w *= sc0;
    aR[0] = v0;
    float4 v1 = *(const float4*)&emb[(size_t)(row0 + rA + 64) * DDIM + d0 + partA * 4];
    v1.x *= sc1; v1.y *= sc1; v1.z *= sc1; v1.w *= sc1;
    aR[1] = v1;
#pragma unroll
    for (int i = 0; i < 8; ++i) {
      int idx = tid + 256 * i;        // 2048 float4s of the B chunk
      int dl  = idx >> 7;             // 0..15
      int k4  = idx & 127;
      bR[i] = *(const float4*)&ctrT[(size_t)(d0 + dl) * KCLS + k4 * 4];
    }
  };
  // commit staged registers to LDS
  auto store_chunk = [&]() {
#pragma unroll
    for (int p = 0; p < 2; ++p)
      *(float4*)&A_lds[(rA + 64 * p) * DC + partA * 4] = aR[p];
#pragma unroll
    for (int i = 0; i < 8; ++i) {
      int idx = tid + 256 * i;
      int dl  = idx >> 7;
      int k4  = idx & 127;
      B_lds[(k4 * 4 + 0) * DC + dl] = bR[i].x;
      B_lds[(k4 * 4 + 1) * DC + dl] = bR[i].y;
      B_lds[(k4 * 4 + 2) * DC + dl] = bR[i].z;
      B_lds[(k4 * 4 + 3) * DC + dl] = bR[i].w;
    }
  };

  load_chunk(0);
  for (int ch = 0; ch < DDIM / DC; ++ch) {
    __syncthreads();                   // all waves done reading previous chunk
    store_chunk();
    __syncthreads();                   // chunk visible in LDS
    if (ch + 1 < DDIM / DC) load_chunk((ch + 1) * DC);  // overlap with WMMAs
#pragma unroll
    for (int s = 0; s < DC / 4; ++s) {
      const int kb = s * 4 + hi * 2;   // K-pair this lane-half holds
      v2f a = *(const v2f*)&A_lds[(wid * 16 + lo) * DC + kb];
#pragma unroll
      for (int t = 0; t < 32; ++t) {
        v2f bf = *(const v2f*)&B_lds[(t * 16 + lo) * DC + kb];
        c[t] = __builtin_amdgcn_wmma_f32_16x16x4_f32(
            false, a, false, bf, (short)0, c[t], false, false);
      }
    }
  }

  // ---- epilogue: per-row own-substitution + log-softmax, all in registers
  const float wv = fmaxf(wptr[0], 1e-6f);
  float lane_loss = 0.0f;

#pragma unroll
  for (int v = 0; v < 8; ++v) {
    const int gr    = row0 + wid * 16 + v + 8 * hi;   // global row this (lane,v) holds
    const int l     = labels[gr];
    const float cnt = counts[l];
    const float sn2 = snorm2[l];
    const int  lt     = l >> 4;
    const bool holder = (lo == (l & 15));
    // extract raw sims[gr][l] (meaningful on holder lane only)
    float vlab = 0.0f;
#pragma unroll
    for (int t = 0; t < 32; ++t) if (t == lt) vlab = c[t][v];
    const float sdot = vlab * cnt;                     // dot(e, sums[l])
    const float nrm2 = fmaxf(sn2 - 2.0f * sdot + 1.0f, 1e-24f);
    const float own  = (sdot - 1.0f) / sqrtf(nrm2);    // exclusive-centroid sim
    const float labval = (cnt > 1.0f) ? own : vlab;
    // row max over all 512 cols (tiles + half-wave shuffle)
    float mx = -3.4e38f;
#pragma unroll
    for (int t = 0; t < 32; ++t) {
      float x = c[t][v];
      if (t == lt && holder) x = labval;
      mx = fmaxf(mx, x);
    }
#pragma unroll
    for (int m = 1; m <= 8; m <<= 1) mx = fmaxf(mx, __shfl_xor(mx, m, 32));
    // sum exp(wv*(x - mx))   (bias cancels in log-softmax)
    float se = 0.0f;
#pragma unroll
    for (int t = 0; t < 32; ++t) {
      float x = c[t][v];
      if (t == lt && holder) x = labval;
      se += __expf(wv * (x - mx));
    }
#pragma unroll
    for (int m = 1; m <= 8; m <<= 1) se += __shfl_xor(se, m, 32);
    if (holder) {
      float lp = wv * (labval - mx) - __logf(se);
      lane_loss += -lp * (1.0f / (float)NROWS);
    }
  }
#pragma unroll
  for (int m = 1; m <= 16; m <<= 1) lane_loss += __shfl_xor(lane_loss, m, 32);
  if (lane == 0) red[wid] = lane_loss;
  __syncthreads();
  if (tid == 0) {
    float t = 0.0f;
#pragma unroll
    for (int w = 0; w < 8; ++w) t += red[w];
    atomicAdd(loss_acc, t);
  }
}

__global__ void k_final(const float* __restrict__ loss_acc, float* __restrict__ out) {
  out[0] = loss_acc[0];
}

// ---------------------------------------------------------------------------
extern "C" void kernel_launch(void* const* d_in, const int* in_sizes, int n_in,
                              void* d_out, int out_size, void* d_ws, size_t ws_size,
                              hipStream_t stream) {
  (void)in_sizes; (void)n_in; (void)out_size; (void)ws_size;
  const float* emb    = (const float*)d_in[0];
  const int*   labels = (const int*)d_in[1];
  const float* wptr   = (const float*)d_in[2];
  const float* bptr   = (const float*)d_in[3];
  // d_in[4] = num_classes (hardcoded KCLS=512)

  float* ws       = (float*)d_ws;
  float* inv_norm = ws;                         // 65536
  float* counts   = inv_norm + NROWS;           // 512
  float* snorm2   = counts + KCLS;              // 512
  float* loss_acc = snorm2 + KCLS;              // 1 (padded to 4)
  float* sums     = loss_acc + 4;               // 512*512
  float* ctrT     = sums + (size_t)KCLS * DDIM; // 512*512

  // zero counts + snorm2 + loss + sums (contiguous)
  int nz = KCLS + KCLS + 4 + KCLS * DDIM;
  k_zero<<<(nz + 255) / 256, 256, 0, stream>>>(counts, nz);
  k_norm_accum<<<NROWS / 8, 256, 0, stream>>>(emb, labels, inv_norm, sums, counts);
  k_centroid<<<KCLS, 256, 0, stream>>>(sums, counts, ctrT, snorm2);
  k_gemm_loss<<<NROWS / 128, 256, 0, stream>>>(emb, labels, inv_norm, ctrT,
                                               counts, snorm2, wptr, bptr, loss_acc);
  k_final<<<1, 1, 0, stream>>>(loss_acc, (float*)d_out);
}